// GNN_72129680769634
// MI455X (gfx1250) — compile-verified
//
#include <hip/hip_runtime.h>
#include <hip/hip_bf16.h>

typedef float v2f __attribute__((ext_vector_type(2)));
typedef float v8f __attribute__((ext_vector_type(8)));

#define DIM 128
#define HEADS 8
#define HEAD_DIM 16
#define LDS_STRIDE 132   // 128 + 4 pad: bank = 4*row + col mod 64, conflict-free

// monotone float<->int encoding so signed-int atomicMax implements float max
__device__ __forceinline__ int fenc(float f) {
  int i = __float_as_int(f);
  return i ^ ((i >> 31) & 0x7FFFFFFF);
}
__device__ __forceinline__ float fdec(int k) {
  return __int_as_float(k ^ ((k >> 31) & 0x7FFFFFFF));
}

__device__ __forceinline__ float leaky02(float v) {
  return v > 0.0f ? v : 0.2f * v;
}

// ---------------------------------------------------------------------------
// Kernel 1: xh = x @ W^T + b_proj   via V_WMMA_F32_16X16X4_F32
// block = 256 threads (8 waves). Block owns a 16-row M tile; wave w owns
// N-cols [16w,16w+16). K=128 in 32 steps of 4.
// ---------------------------------------------------------------------------
__global__ __launch_bounds__(256) void gat_gemm(const float* __restrict__ x,
                                                const float* __restrict__ W,
                                                const float* __restrict__ b_proj,
                                                float* __restrict__ xh, int n) {
  __shared__ float lds_x[16 * LDS_STRIDE];
  const int rowBase = blockIdx.x * 16;
  const int tid = threadIdx.x;

  // stage 16x128 fp32 x-tile into LDS (2 float4 per thread, coalesced)
  for (int i = 0; i < 2; ++i) {
    int idx4 = tid + i * 256;        // 0..511 float4 slots
    int r = idx4 >> 5;               // 32 float4 per row
    int c4 = idx4 & 31;
    int row = rowBase + r;
    if (row >= n) row = n - 1;       // clamp (tail-safe)
    const float4 v = *(const float4*)(x + (size_t)row * DIM + c4 * 4);
    float* p = &lds_x[r * LDS_STRIDE + c4 * 4];
    p[0] = v.x; p[1] = v.y; p[2] = v.z; p[3] = v.w;
  }
  __syncthreads();

  const int wave = tid >> 5;
  const int lane = tid & 31;
  const int colBase = wave * 16;
  const int laneHalf = lane >> 4;    // selects K pair {0,1} vs {2,3}
  const int laneIdx  = lane & 15;    // M for A-frag, N for B-frag

  const float* arow = &lds_x[laneIdx * LDS_STRIDE];
  const float* wrow = W + (size_t)(colBase + laneIdx) * DIM; // B[k][n] = W[n][k]

  v8f acc = {};
  for (int k0 = 0; k0 < DIM; k0 += 4) {
    const int kk = k0 + laneHalf * 2;
    v2f a; a[0] = arow[kk]; a[1] = arow[kk + 1];
    v2f b; b[0] = wrow[kk]; b[1] = wrow[kk + 1];
    acc = __builtin_amdgcn_wmma_f32_16x16x4_f32(
        /*neg_a=*/false, a, /*neg_b=*/false, b,
        /*c_mod=*/(short)0, acc, /*reuse_a=*/false, /*reuse_b=*/false);
  }

  // D layout: VGPR v -> M = v + 8*laneHalf, N = laneIdx
  const int col = colBase + laneIdx;
  const float bp = b_proj[col];
  for (int v = 0; v < 8; ++v) {
    const int m = rowBase + v + laneHalf * 8;
    if (m < n) xh[(size_t)m * DIM + col] = acc[v] + bp;
  }
}

// ---------------------------------------------------------------------------
// Kernel 2: per-(node,head) attention logits
// ---------------------------------------------------------------------------
__global__ void gat_att(const float* __restrict__ xh,
                        const float* __restrict__ att_src,
                        const float* __restrict__ att_dst,
                        float* __restrict__ a_src, float* __restrict__ a_dst,
                        int n) {
  int i = blockIdx.x * blockDim.x + threadIdx.x;
  if (i >= n * HEADS) return;
  const int node = i >> 3, h = i & 7;
  const float* row = xh + (size_t)node * DIM + h * HEAD_DIM;
  const float* as = att_src + h * HEAD_DIM;
  const float* ad = att_dst + h * HEAD_DIM;
  float s = 0.f, d = 0.f;
#pragma unroll
  for (int k = 0; k < HEAD_DIM; ++k) { float v = row[k]; s += v * as[k]; d += v * ad[k]; }
  a_src[i] = s;
  a_dst[i] = d;
}

// ---------------------------------------------------------------------------
// Kernel 3: init out=bias (broadcast), mx=enc(-FLT_MAX), denom=0
// ---------------------------------------------------------------------------
__global__ void gat_init(float* __restrict__ out, int* __restrict__ mx,
                         float* __restrict__ denom, const float* __restrict__ bias,
                         int n) {
  int i = blockIdx.x * blockDim.x + threadIdx.x;
  if (i < n * DIM) out[i] = bias[i & (DIM - 1)];
  if (i < n * HEADS) { mx[i] = fenc(-3.402823466e38f); denom[i] = 0.f; }
}

// ---------------------------------------------------------------------------
// Edge pass 1: segment max of leaky-relu scores (ordered-int atomicMax)
// one thread per (edge, head); edges [E, E+n) are self-loops
// ---------------------------------------------------------------------------
__global__ void gat_edge_max(const int* __restrict__ ei,
                             const float* __restrict__ a_src,
                             const float* __restrict__ a_dst,
                             int* __restrict__ mx, int E, int n) {
  int i = blockIdx.x * blockDim.x + threadIdx.x;
  const int total = (E + n) * HEADS;
  if (i >= total) return;
  const int e = i >> 3, h = i & 7;
  const int s = (e < E) ? ei[e] : (e - E);
  const int d = (e < E) ? ei[E + e] : (e - E);
  const float v = leaky02(a_src[s * HEADS + h] + a_dst[d * HEADS + h]);
  atomicMax(&mx[d * HEADS + h], fenc(v));
}

// ---------------------------------------------------------------------------
// Edge pass 2: denom = segment_sum(exp(alpha - mx))
// ---------------------------------------------------------------------------
__global__ void gat_edge_sum(const int* __restrict__ ei,
                             const float* __restrict__ a_src,
                             const float* __restrict__ a_dst,
                             const int* __restrict__ mx,
                             float* __restrict__ denom, int E, int n) {
  int i = blockIdx.x * blockDim.x + threadIdx.x;
  const int total = (E + n) * HEADS;
  if (i >= total) return;
  const int e = i >> 3, h = i & 7;
  const int s = (e < E) ? ei[e] : (e - E);
  const int d = (e < E) ? ei[E + e] : (e - E);
  const float v = leaky02(a_src[s * HEADS + h] + a_dst[d * HEADS + h]);
  const float ex = __expf(v - fdec(mx[d * HEADS + h]));
  atomicAdd(&denom[d * HEADS + h], ex);
}

// ---------------------------------------------------------------------------
// Edge pass 3: out[dst] += (exp(alpha-mx)/denom) * xh[src]
// one thread per (edge, feature-column t in 0..127); h = t/16
// ---------------------------------------------------------------------------
__global__ void gat_edge_scatter(const int* __restrict__ ei,
                                 const float* __restrict__ a_src,
                                 const float* __restrict__ a_dst,
                                 const int* __restrict__ mx,
                                 const float* __restrict__ denom,
                                 const float* __restrict__ xh,
                                 float* __restrict__ out, int E, int n) {
  int i = blockIdx.x * blockDim.x + threadIdx.x;
  const int total = (E + n) * DIM;          // 217.6M < 2^31
  if (i >= total) return;
  const int e = i >> 7, t = i & (DIM - 1);
  const int h = t >> 4;
  const int s = (e < E) ? ei[e] : (e - E);
  const int d = (e < E) ? ei[E + e] : (e - E);
  const float v = leaky02(a_src[s * HEADS + h] + a_dst[d * HEADS + h]);
  const float w = __expf(v - fdec(mx[d * HEADS + h])) / denom[d * HEADS + h];
  atomicAdd(&out[(size_t)d * DIM + t], w * xh[(size_t)s * DIM + t]);
}

// ---------------------------------------------------------------------------
extern "C" void kernel_launch(void* const* d_in, const int* in_sizes, int n_in,
                              void* d_out, int out_size, void* d_ws, size_t ws_size,
                              hipStream_t stream) {
  const float* x       = (const float*)d_in[0];
  const int*   ei      = (const int*)  d_in[1];
  const float* W       = (const float*)d_in[2];
  const float* b_proj  = (const float*)d_in[3];
  const float* att_src = (const float*)d_in[4];
  const float* att_dst = (const float*)d_in[5];
  const float* bias    = (const float*)d_in[6];
  float* out = (float*)d_out;

  const int n = in_sizes[0] / DIM;   // 100000
  const int E = in_sizes[1] / 2;     // 1600000

  // workspace layout
  float* xh    = (float*)d_ws;                       // n*128
  float* a_src = xh + (size_t)n * DIM;               // n*8
  float* a_dst = a_src + (size_t)n * HEADS;          // n*8
  int*   mx    = (int*)(a_dst + (size_t)n * HEADS);  // n*8
  float* denom = (float*)(mx + (size_t)n * HEADS);   // n*8

  const int nTiles = (n + 15) / 16;
  gat_gemm<<<nTiles, 256, 0, stream>>>(x, W, b_proj, xh, n);

  const int nh = n * HEADS;
  gat_att<<<(nh + 255) / 256, 256, 0, stream>>>(xh, att_src, att_dst, a_src, a_dst, n);

  gat_init<<<(n * DIM + 255) / 256, 256, 0, stream>>>(out, mx, denom, bias, n);

  const int eh = (E + n) * HEADS;
  gat_edge_max<<<(eh + 255) / 256, 256, 0, stream>>>(ei, a_src, a_dst, mx, E, n);
  gat_edge_sum<<<(eh + 255) / 256, 256, 0, stream>>>(ei, a_src, a_dst, mx, denom, E, n);

  const long long et = (long long)(E + n) * DIM;
  gat_edge_scatter<<<(int)((et + 255) / 256), 256, 0, stream>>>(
      ei, a_src, a_dst, mx, denom, xh, out, E, n);
}